// BagOfWords_37941741093587
// MI455X (gfx1250) — compile-verified
//
#include <hip/hip_runtime.h>

#define NTOK    1100
#define SEQ     200
#define ROWS    4
#define CSTRIDE 1104          // padded counter row stride (dwords): uint4-aligned, shifts banks per row
#define BLOCK   256           // 8 wave32 waves

__global__ __launch_bounds__(BLOCK)
void bow_hist_kernel(const int* __restrict__ in, float* __restrict__ out, int batch)
{
    __shared__ __align__(16) unsigned cnt[ROWS * CSTRIDE];  // 17664 B
    __shared__ __align__(16) int      toks[ROWS * SEQ];     // 3200 B

    const int tid  = threadIdx.x;
    const int row0 = blockIdx.x * ROWS;
    const int nrow = min(ROWS, batch - row0);

    // ---- 1) kick off async global->LDS copy of this block's tokens ----------
    // 16 B (4 tokens) per active lane; CDNA5 async-copy path, tracked by ASYNCcnt.
    {
        const int nvec = nrow * (SEQ / 4);          // number of b128 transfers
        if (tid < nvec) {
            // low 32 bits of a generic pointer to LDS == LDS byte offset
            unsigned lds_dst = (unsigned)(size_t)(const void*)toks + (unsigned)tid * 16u;
            unsigned goff    = (unsigned)tid * 16u; // byte offset from row base
            const int* src   = in + (size_t)row0 * SEQ;   // wave-uniform -> SGPR pair
            asm volatile("global_load_async_to_lds_b128 %0, %1, %2"
                         :
                         : "v"(lds_dst), "v"(goff), "s"(src)
                         : "memory");
        }
    }

    // ---- 2) zero counters while the async copy is in flight ------------------
    {
        uint4* c4 = (uint4*)cnt;
        const int n4 = (ROWS * CSTRIDE) / 4;        // 1104 uint4 stores
        for (int i = tid; i < n4; i += BLOCK)
            c4[i] = make_uint4(0u, 0u, 0u, 0u);
    }

    asm volatile("s_wait_asynccnt 0" ::: "memory");
    __syncthreads();

    // ---- 3) LDS histogram via ds_add_u32 -------------------------------------
    {
        const int n = nrow * SEQ;
        for (int i = tid; i < n; i += BLOCK) {
            int r   = i / SEQ;                      // constant div -> mulhi+shift
            int tok = toks[i];                      // in [0, 1100)
            atomicAdd(&cnt[r * CSTRIDE + tok], 1u); // non-returning ds_add_u32
        }
    }
    __syncthreads();

    // ---- 4) convert + coalesced store, dropping token id 0 -------------------
    for (int r = 0; r < nrow; ++r) {
        const unsigned* crow = cnt + r * CSTRIDE;
        float* orow = out + (size_t)(row0 + r) * (NTOK - 1);
        for (int c = 1 + tid; c < NTOK; c += BLOCK)
            orow[c - 1] = (float)crow[c];
    }
}

extern "C" void kernel_launch(void* const* d_in, const int* in_sizes, int n_in,
                              void* d_out, int out_size, void* d_ws, size_t ws_size,
                              hipStream_t stream)
{
    const int* in  = (const int*)d_in[0];
    float*     out = (float*)d_out;
    const int batch = in_sizes[0] / SEQ;            // 16384
    const int grid  = (batch + ROWS - 1) / ROWS;    // 4096 workgroups
    hipLaunchKernelGGL(bow_hist_kernel, dim3(grid), dim3(BLOCK), 0, stream,
                       in, out, batch);
}